// PoolingLayer_58171037057498
// MI455X (gfx1250) — compile-verified
//
#include <hip/hip_runtime.h>
#include <math.h>

typedef __attribute__((ext_vector_type(2))) float v2f;
typedef __attribute__((ext_vector_type(4))) float v4f;
typedef __attribute__((ext_vector_type(8))) float v8f;

struct TopkState {
    unsigned long long prefix;   // accumulated radix-select threshold (packed key)
    unsigned int remaining;      // rank still being searched within current candidates
    unsigned int selCount;       // compaction counter
    float scale;                 // 1 / ||v||
};

// Orderable transform: u ascending <=> float ascending; dkey = ~u ascending <=> float DESCENDING.
__device__ __forceinline__ unsigned long long packKey(float val, unsigned int idx) {
    unsigned int ub = __float_as_uint(val);
    unsigned int u  = ub ^ ((ub >> 31) ? 0xFFFFFFFFu : 0x80000000u);
    return (((unsigned long long)(~u)) << 32) | (unsigned long long)idx;
}

// ---------------- Kernel 1: ||v|| reduction + state/bins init ----------------
__global__ void init_norm_kernel(const float* __restrict__ v, int d, int k,
                                 unsigned int* __restrict__ bins, TopkState* __restrict__ st) {
    __shared__ float red[256];
    int t = threadIdx.x;
    float s = 0.f;
    for (int i = t; i < d; i += 256) { float e = v[i]; s += e * e; }
    red[t] = s;
    __syncthreads();
    for (int off = 128; off > 0; off >>= 1) {
        if (t < off) red[t] += red[t + off];
        __syncthreads();
    }
    bins[t] = 0u;  // 256 bins, 256 threads
    if (t == 0) {
        st->prefix    = 0ull;
        st->remaining = (unsigned int)k;
        st->selCount  = 0u;
        st->scale     = 1.0f / sqrtf(red[0]);
    }
}

// ---------------- Kernel 2: GEMV via V_WMMA_F32_16X16X4_F32 ----------------
// One wave computes scores for 16 rows. A-tile layout (ISA 7.12.2, 32-bit A 16x4):
// lanes 0-15 hold tile K=0,1; lanes 16-31 hold tile K=2,3; M = lane % 16.
// We assign global k's: low lanes take {k0,k0+1} (wmma0) / {k0+2,k0+3} (wmma1),
// high lanes take {k0+4,k0+5} / {k0+6,k0+7}; B carries v at matching k's,
// replicated across all 16 columns, so D[m][n] = dot for every n.
__global__ void gemv_wmma_kernel(const float* __restrict__ x, const float* __restrict__ v,
                                 float* __restrict__ y, int N, int d,
                                 const TopkState* __restrict__ st) {
    int lane    = threadIdx.x & 31;
    int wave    = blockIdx.x * (blockDim.x >> 5) + (threadIdx.x >> 5);
    int rowBase = wave * 16;
    if (rowBase >= N) return;                 // wave-uniform exit (EXEC all-ones for WMMA)
    int half = lane >> 4;                     // 0: lanes 0-15, 1: lanes 16-31
    int m    = lane & 15;

    const float* xr = x + (size_t)(rowBase + m) * (size_t)d + (size_t)(4 * half);
    const float* vr = v + 4 * half;

    v8f c = {};
    for (int k0 = 0; k0 < d; k0 += 8) {
        v4f ax = *(const v4f*)(xr + k0);      // b128: this lane's 4 k-values of its row
        v4f bv = *(const v4f*)(vr + k0);      // b128: matching 4 v-values (broadcast per half)
        v2f a0 = { ax.x, ax.y }, b0 = { bv.x, bv.y };
        v2f a1 = { ax.z, ax.w }, b1 = { bv.z, bv.w };
        c = __builtin_amdgcn_wmma_f32_16x16x4_f32(false, a0, false, b0, (short)0, c, false, false);
        c = __builtin_amdgcn_wmma_f32_16x16x4_f32(false, a1, false, b1, (short)0, c, false, false);
    }

    float scale = st->scale;
    // C/D layout: VGPR r, lanes 0-15 -> M=r; lanes 16-31 -> M=8+r. All columns equal.
    if (lane == 0) {
        #pragma unroll
        for (int r = 0; r < 8; ++r) y[rowBase + r] = c[r] * scale;
    } else if (lane == 16) {
        #pragma unroll
        for (int r = 0; r < 8; ++r) y[rowBase + 8 + r] = c[r] * scale;
    }
}

// ---------------- Kernel 3: radix-select histogram over one byte ----------------
__global__ void hist_kernel(const float* __restrict__ y, int N, int bytePos,
                            unsigned int* __restrict__ bins, const TopkState* __restrict__ st) {
    __shared__ unsigned int lb[256];
    int t = threadIdx.x;
    if (t < 256) lb[t] = 0u;
    __syncthreads();
    unsigned long long prefix = st->prefix;
    int shiftHi = (bytePos + 1) * 8;
    for (int i = blockIdx.x * blockDim.x + t; i < N; i += gridDim.x * blockDim.x) {
        unsigned long long packed = packKey(y[i], (unsigned int)i);
        bool cand = (bytePos == 7) || ((packed >> shiftHi) == (prefix >> shiftHi));
        if (cand) {
            unsigned int b = (unsigned int)((packed >> (bytePos * 8)) & 0xFFull);
            atomicAdd(&lb[b], 1u);
        }
    }
    __syncthreads();
    if (t < 256 && lb[t]) atomicAdd(&bins[t], lb[t]);
}

// ---------------- Kernel 4: pick byte, update prefix, zero bins ----------------
__global__ void scan_kernel(unsigned int* __restrict__ bins, TopkState* __restrict__ st,
                            int bytePos) {
    unsigned int rem = st->remaining;
    unsigned int cum = 0;
    int chosen = 255;
    bool found = false;
    for (int j = 0; j < 256; ++j) {
        unsigned int c = bins[j];
        if (!found && cum + c >= rem) { found = true; chosen = j; st->remaining = rem - cum; }
        if (!found) cum += c;
        bins[j] = 0u;   // ready for next pass
    }
    st->prefix |= ((unsigned long long)chosen) << (bytePos * 8);
}

// ---------------- Kernel 5: compact exactly-k selected keys ----------------
__global__ void compact_kernel(const float* __restrict__ y, int N,
                               unsigned long long* __restrict__ sel, TopkState* __restrict__ st) {
    unsigned long long T = st->prefix;   // k-th smallest packed key (keys unique)
    for (int i = blockIdx.x * blockDim.x + threadIdx.x; i < N; i += gridDim.x * blockDim.x) {
        unsigned long long packed = packKey(y[i], (unsigned int)i);
        if (packed <= T) {
            unsigned int p = atomicAdd(&st->selCount, 1u);
            sel[p] = packed;
        }
    }
}

// ---------------- Kernel 6: bitonic sort (128KB LDS) + gate + gather ----------------
__global__ void sort_output_kernel(const float* __restrict__ x, int d,
                                   const unsigned long long* __restrict__ sel,
                                   float* __restrict__ out, int k) {
    extern __shared__ unsigned long long s[];
    int t = threadIdx.x, nt = blockDim.x;
    for (int i = t; i < k; i += nt) s[i] = sel[i];
    __syncthreads();
    for (int len = 2; len <= k; len <<= 1) {
        for (int j = len >> 1; j > 0; j >>= 1) {
            for (int i = t; i < k; i += nt) {
                int ixj = i ^ j;
                if (ixj > i) {
                    unsigned long long a = s[i], b = s[ixj];
                    bool asc = ((i & len) == 0);
                    if ((a > b) == asc) { s[i] = b; s[ixj] = a; }
                }
            }
            __syncthreads();
        }
    }
    // ascending packed key == descending score; unpack, gate, gather x[idx,0]
    for (int i = t; i < k; i += nt) {
        unsigned long long p = s[i];
        unsigned int idx  = (unsigned int)(p & 0xFFFFFFFFull);
        unsigned int u    = ~((unsigned int)(p >> 32));
        unsigned int bits = (u >> 31) ? (u ^ 0x80000000u) : ~u;
        float val  = __uint_as_float(bits);
        float gate = 1.0f / (1.0f + expf(-val));
        out[i] = x[(size_t)idx * (size_t)d] * gate;
    }
}

extern "C" void kernel_launch(void* const* d_in, const int* in_sizes, int n_in,
                              void* d_out, int out_size, void* d_ws, size_t ws_size,
                              hipStream_t stream) {
    (void)n_in; (void)ws_size;
    const int d = in_sizes[1];            // 512
    const int N = in_sizes[0] / d;        // 200000
    const int k = out_size;               // 16384 (power of two; bitonic relies on this)

    const float* x = (const float*)d_in[0];
    const float* v = (const float*)d_in[1];
    float* out = (float*)d_out;

    // Workspace layout (~935 KB total)
    char* ws = (char*)d_ws;
    float* y = (float*)ws;                                     // N scores
    size_t yBytes   = (((size_t)N * 4) + 255) & ~(size_t)255;
    unsigned long long* sel = (unsigned long long*)(ws + yBytes);  // k packed keys
    size_t selBytes = (((size_t)k * 8) + 255) & ~(size_t)255;
    unsigned int* bins = (unsigned int*)(ws + yBytes + selBytes);  // 256 bins
    TopkState* st = (TopkState*)(ws + yBytes + selBytes + 1024);

    init_norm_kernel<<<1, 256, 0, stream>>>(v, d, k, bins, st);

    int waves  = (N + 15) / 16;           // one wave per 16 rows
    int blocks = (waves + 7) / 8;         // 256 threads = 8 waves per block
    gemv_wmma_kernel<<<blocks, 256, 0, stream>>>(x, v, y, N, d, st);

    for (int p = 0; p < 8; ++p) {
        int bp = 7 - p;                   // MSB byte first
        hist_kernel<<<208, 256, 0, stream>>>(y, N, bp, bins, st);
        scan_kernel<<<1, 1, 0, stream>>>(bins, st, bp);
    }

    compact_kernel<<<208, 256, 0, stream>>>(y, N, sel, st);

    sort_output_kernel<<<1, 1024, (size_t)k * sizeof(unsigned long long), stream>>>(
        x, d, sel, out, k);
}